// HPCrnn_67542655696945
// MI455X (gfx1250) — compile-verified
//
#include <hip/hip_runtime.h>
#include <hip/hip_bf16.h>
#include <math.h>

typedef float v2f __attribute__((ext_vector_type(2)));
typedef float v8f __attribute__((ext_vector_type(8)));

#define ECN    100
#define CA1N   100
#define CA3N   100
#define TLEN   100
#define BSZ    4096
#define TILE_B 32              // batch rows per workgroup (2 waves x 16)
#define NWAVES 2
#define NTHREADS (NWAVES * 32)
#define S      114             // state row stride in floats (even, 114%64=50 -> conflict-free b64)
#define WPS    112             // weight pair-row column count (2*112 % 64 == 32 -> halves disjoint)
#define NT     7               // 7 n-tiles of 16 cover 100 (pad to 112)

__device__ __forceinline__ float sigmoidf_(float x) {
    return __builtin_amdgcn_rcpf(1.0f + __expf(-x));
}

// drive[t][c] = sum_j exp(-0.5*((t - j*100/99)/5)^2) * Wca3ca1[j][c]
__global__ void HPCrnn_drive_kernel(const float* __restrict__ Wca3ca1,
                                    float* __restrict__ drive) {
    __shared__ float g[CA3N];
    const int t = blockIdx.x;
    const int c = threadIdx.x;
    if (c < CA3N) {
        float center = (float)c * (100.0f / 99.0f);
        float d = ((float)t - center) * (1.0f / 5.0f);
        g[c] = __expf(-0.5f * d * d);
    }
    __syncthreads();
    if (c < CA1N) {
        float s = 0.0f;
        for (int j = 0; j < CA3N; ++j)
            s += g[j] * Wca3ca1[j * CA1N + c];
        drive[t * CA1N + c] = s;
    }
}

__global__ __launch_bounds__(NTHREADS)
void HPCrnn_scan_kernel(const int* __restrict__ cue_train,
                        const float* __restrict__ ec3_last,
                        const float* __restrict__ ec5_last,
                        const float* __restrict__ cueL,
                        const float* __restrict__ cueR,
                        const float* __restrict__ Wec3ca1,
                        const float* __restrict__ Wca1ec5,
                        const float* __restrict__ Wca1act,
                        const float* __restrict__ ca1bias,
                        const float* __restrict__ drive,
                        float* __restrict__ out) {
    extern __shared__ float smem[];
    // weights stored K-pair-interleaved: element (k, col) at ((k>>1)*WPS + col)*2 + (k&1)
    float* sW1   = smem;                       // [50][WPS][2]  Wec3ca1 (K=ec, N=ca1)
    float* sW2   = sW1  + (ECN / 2) * WPS * 2; // [50][WPS][2]  Wca1ec5 (K=ca1, N=ec)
    float* sEc3  = sW2  + (CA1N / 2) * WPS * 2;// [TILE_B][S]
    float* sCa1  = sEc3 + TILE_B * S;          // [TILE_B][S]
    float* sCueL = sCa1 + TILE_B * S;          // [S]
    float* sCueR = sCueL + S;                  // [S]
    float* sBias = sCueR + S;                  // [S]
    int*   sCue  = (int*)(sBias + S);          // [TILE_B][TLEN]

    const int tid  = threadIdx.x;
    const int lane = tid & 31;
    const int wv   = tid >> 5;
    const int lh   = lane & 15;
    const int hf   = lane >> 4;            // half-wave select
    const int m0   = wv * 16;              // this wave's 16-row tile inside the WG
    const int B0   = blockIdx.x * TILE_B;  // global batch base of WG

    // ---------------- cooperative LDS init ----------------
    for (int i = tid; i < ECN * WPS; i += NTHREADS) {
        int k = i / WPS, col = i % WPS;
        int idx = ((k >> 1) * WPS + col) * 2 + (k & 1);
        sW1[idx] = (col < CA1N) ? Wec3ca1[k * CA1N + col] : 0.0f;
        sW2[idx] = (col < ECN)  ? Wca1ec5[k * ECN + col]  : 0.0f;
    }
    for (int i = tid; i < TILE_B * S; i += NTHREADS) {
        int r = i / S, c = i % S;
        sEc3[i] = (c < ECN) ? ec3_last[(size_t)(B0 + r) * ECN + c] : 0.0f;
        sCa1[i] = 0.0f;
    }
    for (int i = tid; i < S; i += NTHREADS) {
        sCueL[i] = (i < ECN)  ? cueL[i]    : 0.0f;
        sCueR[i] = (i < ECN)  ? cueR[i]    : 0.0f;
        sBias[i] = (i < CA1N) ? ca1bias[i] : 0.0f;
    }
    for (int i = tid; i < TILE_B * TLEN; i += NTHREADS) {
        int r = i / TLEN, t = i % TLEN;
        sCue[i] = cue_train[(size_t)(B0 + r) * TLEN + t];
    }

    // ---------------- register-resident carries (C/D fragment layout) ----------------
    v8f ec3r[NT], ec5r[NT], acc[NT];
    #pragma unroll
    for (int n = 0; n < NT; ++n) {
        const int c  = n * 16 + lh;
        const bool cv = (c < ECN);
        #pragma unroll
        for (int j = 0; j < 8; ++j) {
            const int r = m0 + j + 8 * hf;
            ec3r[n][j] = cv ? ec3_last[(size_t)(B0 + r) * ECN + c] : 0.0f;
            ec5r[n][j] = cv ? ec5_last[(size_t)(B0 + r) * ECN + c] : 0.0f;
            acc[n][j]  = 0.0f;
        }
    }
    __syncthreads();

    // output segment pointers (flat concat: actCell, ca1his, ec3, ec5, ca1)
    float* outAct = out;
    float* outHis = out + (size_t)BSZ * 2;
    float* outEc3 = outHis + (size_t)TLEN * BSZ * CA1N;
    float* outEc5 = outEc3 + (size_t)BSZ * ECN;
    float* outCa1 = outEc5 + (size_t)BSZ * ECN;

    const v8f vzero = {0.f, 0.f, 0.f, 0.f, 0.f, 0.f, 0.f, 0.f};

    #pragma unroll 1
    for (int t = 0; t < TLEN; ++t) {
        // ---------- GEMM1: acc = ec3_tile[16x100] @ W1[100x112] ----------
        #pragma unroll
        for (int n = 0; n < NT; ++n) acc[n] = vzero;

        #pragma unroll
        for (int k0 = 0; k0 < ECN; k0 += 4) {
            const int kp = (k0 >> 1) + hf;                     // K-pair row for this half-wave
            v2f a = *(const v2f*)&sEc3[(m0 + lh) * S + k0 + 2 * hf];
            #pragma unroll
            for (int n = 0; n < NT; ++n) {
                v2f b = *(const v2f*)&sW1[(kp * WPS + n * 16 + lh) * 2];  // (W[kk][c], W[kk+1][c])
                acc[n] = __builtin_amdgcn_wmma_f32_16x16x4_f32(
                    false, a, false, b, (short)0, acc[n], false, false);
            }
        }

        // ---------- epilogue 1: ca1 = relu(drive*(1+sigmoid(x)) - bias) ----------
        const size_t hisBase = ((size_t)t * BSZ + B0) * CA1N;
        #pragma unroll
        for (int n = 0; n < NT; ++n) {
            const int c  = n * 16 + lh;
            const bool cv = (c < CA1N);
            const float dv = cv ? drive[t * CA1N + c] : 0.0f;
            const float bv = sBias[c];
            #pragma unroll
            for (int j = 0; j < 8; ++j) {
                const int r = m0 + j + 8 * hf;          // C/D layout: v_j -> M=j (+8 on hi half)
                float v = fmaxf(dv * (1.0f + sigmoidf_(acc[n][j])) - bv, 0.0f);
                acc[n][j] = v;                          // acc now holds ca1 (kept for outputs)
                if (cv) {
                    sCa1[r * S + c] = v;
                    outHis[hisBase + (size_t)r * CA1N + c] = v;
                }
            }
        }

        // ---------- GEMM2: acc2 = ca1_tile[16x100] @ W2[100x112] ----------
        v8f acc2[NT];
        #pragma unroll
        for (int n = 0; n < NT; ++n) acc2[n] = vzero;

        #pragma unroll
        for (int k0 = 0; k0 < CA1N; k0 += 4) {
            const int kp = (k0 >> 1) + hf;
            v2f a = *(const v2f*)&sCa1[(m0 + lh) * S + k0 + 2 * hf];
            #pragma unroll
            for (int n = 0; n < NT; ++n) {
                v2f b = *(const v2f*)&sW2[(kp * WPS + n * 16 + lh) * 2];
                acc2[n] = __builtin_amdgcn_wmma_f32_16x16x4_f32(
                    false, a, false, b, (short)0, acc2[n], false, false);
            }
        }

        // ---------- epilogue 2: ec5 squash (regs), ec3 decay + cue (regs + LDS store) ----------
        int cuev[8];
        #pragma unroll
        for (int j = 0; j < 8; ++j)
            cuev[j] = sCue[(m0 + j + 8 * hf) * TLEN + t];

        #pragma unroll
        for (int n = 0; n < NT; ++n) {
            const int c  = n * 16 + lh;
            const bool cv = (c < ECN);
            const float Lc = sCueL[c];
            const float Rc = sCueR[c];
            #pragma unroll
            for (int j = 0; j < 8; ++j) {
                const int r = m0 + j + 8 * hf;
                float e5 = ec5r[n][j] + acc2[n][j];
                e5 = 0.69f + 0.3f * sigmoidf_(4.0f * (e5 - 0.3f));
                ec5r[n][j] = e5;
                const float sv = (cuev[j] == 1) ? Lc : ((cuev[j] == -1) ? Rc : 0.0f);
                const float e3 = e5 * ec3r[n][j] + sv;
                ec3r[n][j] = e3;
                if (cv) sEc3[r * S + c] = e3;           // only LDS write; no read-back
            }
        }
    }
    __syncthreads();   // actCell reads sCa1 rows across both waves

    // ---------- actCell = ca1_last @ Wca1act : one (row, act) per thread ----------
    {
        const int r = tid >> 1;
        const int a = tid & 1;
        float s = 0.0f;
        for (int c = 0; c < CA1N; ++c)
            s += sCa1[r * S + c] * Wca1act[c * 2 + a];
        outAct[(size_t)(B0 + r) * 2 + a] = s;
    }

    // ---------- final carries straight from registers ----------
    #pragma unroll
    for (int n = 0; n < NT; ++n) {
        const int c  = n * 16 + lh;
        if (c < ECN) {
            #pragma unroll
            for (int j = 0; j < 8; ++j) {
                const int r = m0 + j + 8 * hf;
                const size_t g = (size_t)(B0 + r) * ECN + c;
                outEc3[g] = ec3r[n][j];
                outEc5[g] = ec5r[n][j];
                outCa1[g] = acc[n][j];     // acc holds last-step ca1
            }
        }
    }
}

extern "C" void kernel_launch(void* const* d_in, const int* in_sizes, int n_in,
                              void* d_out, int out_size, void* d_ws, size_t ws_size,
                              hipStream_t stream) {
    // setup_inputs order:
    // 0 bs, 1 cue_train(i32), 2 ec3_last, 3 ec5_last, 4 ca1_last(unused),
    // 5 cueL, 6 cueR, 7 Wec3ca1, 8 Wca3ca1, 9 Wca1ec5, 10 Wca1act, 11 ca1bias
    const int*   cue_train = (const int*)d_in[1];
    const float* ec3_last  = (const float*)d_in[2];
    const float* ec5_last  = (const float*)d_in[3];
    const float* cueL      = (const float*)d_in[5];
    const float* cueR      = (const float*)d_in[6];
    const float* Wec3ca1   = (const float*)d_in[7];
    const float* Wca3ca1   = (const float*)d_in[8];
    const float* Wca1ec5   = (const float*)d_in[9];
    const float* Wca1act   = (const float*)d_in[10];
    const float* ca1bias   = (const float*)d_in[11];

    float* drive = (float*)d_ws;           // [TLEN][CA1N] scratch
    float* out   = (float*)d_out;

    HPCrnn_drive_kernel<<<TLEN, 128, 0, stream>>>(Wca3ca1, drive);

    const size_t lds = (size_t)(2 * (ECN / 2) * WPS * 2   // sW1 + sW2 (pair-interleaved)
                              + 2 * TILE_B * S            // sEc3 + sCa1
                              + 3 * S) * sizeof(float)    // cueL/cueR/bias
                     + (size_t)(TILE_B * TLEN) * sizeof(int);  // sCue  (~133 KB)

    HPCrnn_scan_kernel<<<BSZ / TILE_B, NTHREADS, lds, stream>>>(
        cue_train, ec3_last, ec5_last, cueL, cueR,
        Wec3ca1, Wca1ec5, Wca1act, ca1bias, drive, out);
}